// GaussianRBF_83330955477629
// MI455X (gfx1250) — compile-verified
//
#include <hip/hip_runtime.h>

typedef __attribute__((ext_vector_type(2))) float v2f;
typedef __attribute__((ext_vector_type(8))) float v8f;

#define O_DIM   1024
#define N_ROWS  32768

// One wave32 computes a 16-row strip of the (N x O) output using
// V_WMMA_F32_16X16X4_F32:  D = A(16x4) * B(4x16) + C
//   A = x rows (D=3 padded to K=4)
//   B = mu columns pre-scaled by 2/sigma^2   (scale is per-column == per-lane)
//   C = -(|x|^2 + |mu|^2)/sigma^2 bias       (column terms are per-lane too)
// so D = -(|x-mu|/sigma)^2 and y = exp(D).
__global__ __launch_bounds__(256) void GaussianRBF_83330955477629_kernel(
    const float* __restrict__ x,        // (N, 3)
    const float* __restrict__ mus,      // (O, 3)
    const float* __restrict__ lsig,     // (O,)
    float* __restrict__ out)            // (N, O)
{
    const int lane = threadIdx.x & 31;
    const int wave = threadIdx.x >> 5;
    const int n0   = (blockIdx.x * 8 + wave) * 16;   // first x-row of this strip
    const int half = lane >> 4;                      // 0: lanes 0-15, 1: lanes 16-31
    const int nl   = lane & 15;                      // column index inside tile

    // ---- A tile (16x4, K padded): lane holds row (n0+nl), K base = 2*half ----
    const int arow = n0 + nl;
    const int kb   = 2 * half;                       // 0 or 2
    v2f a;
    a.x = x[arow * 3 + kb];                          // K = kb   (< 3 always)
    a.y = (kb + 1 < 3) ? x[arow * 3 + kb + 1] : 0.0f; // K = kb+1 (pad lane>=16)

    // ---- |x|^2 for the C-layout rows this lane owns: M = r + 8*half ----
    float xsq8[8];
#pragma unroll
    for (int r = 0; r < 8; ++r) {
        const int row = n0 + r + 8 * half;
        const float x0 = x[row * 3 + 0];
        const float x1 = x[row * 3 + 1];
        const float x2 = x[row * 3 + 2];
        xsq8[r] = x0 * x0 + x1 * x1 + x2 * x2;
    }

    // ================= pass 1: accumulate per-row sums =================
    float rowsum[8];
#pragma unroll
    for (int r = 0; r < 8; ++r) rowsum[r] = 0.0f;

    for (int o0 = 0; o0 < O_DIM; o0 += 16) {
        const int o = o0 + nl;                       // this lane's center column
        const float m0 = mus[o * 3 + 0];
        const float m1 = mus[o * 3 + 1];
        const float m2 = mus[o * 3 + 2];
        const float inv_sig2 = __expf(-2.0f * lsig[o]);   // 1/sigma^2
        const float musq = m0 * m0 + m1 * m1 + m2 * m2;
        const float s = 2.0f * inv_sig2;

        v2f b;                                       // B column scaled by 2/sig^2
        b.x = (half == 0 ? m0 : m2) * s;             // K = kb
        b.y = (half == 0 ? m1 : 0.0f) * s;           // K = kb+1 (pad)

        v8f c;
#pragma unroll
        for (int r = 0; r < 8; ++r)
            c[r] = -(xsq8[r] + musq) * inv_sig2;     // bias so D = -t

        v8f d = __builtin_amdgcn_wmma_f32_16x16x4_f32(
            false, a, false, b, (short)0, c, false, false);

#pragma unroll
        for (int r = 0; r < 8; ++r)
            rowsum[r] += __expf(d[r]);
    }

    // ---- reduce row sums across the 16 lanes of each half (columns) ----
#pragma unroll
    for (int r = 0; r < 8; ++r) {
        float v = rowsum[r];
        v += __shfl_xor(v, 1, 32);
        v += __shfl_xor(v, 2, 32);
        v += __shfl_xor(v, 4, 32);
        v += __shfl_xor(v, 8, 32);                   // masks < 16: stays in half
        rowsum[r] = 1.0f / v;                        // inverse sum, per row
    }

    // ============ pass 2: recompute, normalize, stream to HBM ============
    for (int o0 = 0; o0 < O_DIM; o0 += 16) {
        const int o = o0 + nl;
        const float m0 = mus[o * 3 + 0];
        const float m1 = mus[o * 3 + 1];
        const float m2 = mus[o * 3 + 2];
        const float inv_sig2 = __expf(-2.0f * lsig[o]);
        const float musq = m0 * m0 + m1 * m1 + m2 * m2;
        const float s = 2.0f * inv_sig2;

        v2f b;
        b.x = (half == 0 ? m0 : m2) * s;
        b.y = (half == 0 ? m1 : 0.0f) * s;

        v8f c;
#pragma unroll
        for (int r = 0; r < 8; ++r)
            c[r] = -(xsq8[r] + musq) * inv_sig2;

        v8f d = __builtin_amdgcn_wmma_f32_16x16x4_f32(
            false, a, false, b, (short)0, c, false, false);

#pragma unroll
        for (int r = 0; r < 8; ++r) {
            const size_t row = (size_t)(n0 + r + 8 * half);
            out[row * O_DIM + (size_t)(o0 + nl)] = __expf(d[r]) * rowsum[r];
        }
    }
}

extern "C" void kernel_launch(void* const* d_in, const int* in_sizes, int n_in,
                              void* d_out, int out_size, void* d_ws, size_t ws_size,
                              hipStream_t stream) {
    (void)in_sizes; (void)n_in; (void)out_size; (void)d_ws; (void)ws_size;
    const float* x    = (const float*)d_in[0];   // (8,4096,3)  -> (32768,3)
    const float* mus  = (const float*)d_in[1];   // (1024,3)
    const float* lsig = (const float*)d_in[2];   // (1024,)
    float* out = (float*)d_out;                  // (32768,1024)

    // 32768 rows / (8 waves * 16 rows) = 256 blocks of 256 threads (8 wave32)
    dim3 grid(N_ROWS / (8 * 16));
    dim3 block(256);
    hipLaunchKernelGGL(GaussianRBF_83330955477629_kernel, grid, block, 0, stream,
                       x, mus, lsig, out);
}